// MambaIRSTD_83897891160305
// MI455X (gfx1250) — compile-verified
//
#include <hip/hip_runtime.h>
#include <hip/hip_bf16.h>

typedef __attribute__((ext_vector_type(2))) float v2f;
typedef __attribute__((ext_vector_type(8))) float v8f;

constexpr int DMODEL = 96;
constexpr int DSTATE = 16;
constexpr int DINNER = 192;   // 2 * 96
constexpr int DTRANK = 6;     // ceil(96/16)
constexpr int KDIRS  = 4;
constexpr int BSZ    = 2;
constexpr int HDIM   = 64;
constexpr int WDIM   = 64;
constexpr int LTOK   = HDIM * WDIM;   // 4096
constexpr int CPAD   = 48;            // 38 rows of x_dbl padded to 3 WMMA tiles

// ---------------------------------------------------------------------------
// Generic C[M,N] = A[M,K] * W[N,K]^T  (fp32, V_WMMA_F32_16X16X4_F32)
// One wave computes one 16x16 tile of C. M,N multiples of 16; K multiple of 4.
// ---------------------------------------------------------------------------
__global__ void gemm_xw(const float* __restrict__ A, const float* __restrict__ W,
                        float* __restrict__ C, int M, int N, int Kc) {
    int gtid = blockIdx.x * blockDim.x + threadIdx.x;
    int wave = gtid >> 5;
    int lane = gtid & 31;
    int ntiles = N >> 4;
    int mtiles = M >> 4;
    if (wave >= ntiles * mtiles) return;
    int mt = wave / ntiles;
    int nt = wave % ntiles;

    int half = lane >> 4;       // 0: lanes 0-15, 1: lanes 16-31
    int l16  = lane & 15;

    const float* arow = A + (size_t)(mt * 16 + l16) * Kc;  // row m = mt*16+l16
    const float* wrow = W + (size_t)(nt * 16 + l16) * Kc;  // row n of W (B[k][n]=W[n][k])

    v8f c = {};
    for (int kk = 0; kk < Kc; kk += 4) {
        int k0 = kk + 2 * half;          // per-lane K index (ISA A/B f32 layout)
        v2f a, b;
        a[0] = arow[k0];     a[1] = arow[k0 + 1];
        b[0] = wrow[k0];     b[1] = wrow[k0 + 1];
        c = __builtin_amdgcn_wmma_f32_16x16x4_f32(false, a, false, b,
                                                  (short)0, c, false, false);
    }
    // C/D layout: c[j] -> row mt*16 + j + 8*half, col nt*16 + l16
    float* cc = C + (size_t)(nt * 16 + l16);
#pragma unroll
    for (int j = 0; j < 8; ++j)
        cc[(size_t)(mt * 16 + j + 8 * half) * N] = c[j];
}

// ---------------------------------------------------------------------------
// Depthwise 3x3 conv (SAME, zero pad) + bias + SiLU, scattered into the four
// scan orderings: xs[b][dir][c][pos]
// ---------------------------------------------------------------------------
__global__ void conv_silu_scatter(const float* __restrict__ xz,
                                  const float* __restrict__ cw,
                                  const float* __restrict__ cb,
                                  float* __restrict__ xs) {
    int idx = blockIdx.x * blockDim.x + threadIdx.x;
    if (idx >= BSZ * LTOK * DINNER) return;
    int c = idx % DINNER;
    int t = idx / DINNER;
    int l = t % LTOK;
    int b = t / LTOK;
    int h0 = l / WDIM, w0 = l % WDIM;

    float acc = cb[c];
#pragma unroll
    for (int dh = -1; dh <= 1; ++dh) {
        int hh = h0 + dh;
        if (hh < 0 || hh >= HDIM) continue;
#pragma unroll
        for (int dw = -1; dw <= 1; ++dw) {
            int ww = w0 + dw;
            if (ww < 0 || ww >= WDIM) continue;
            acc += xz[((size_t)(b * LTOK + hh * WDIM + ww)) * (2 * DINNER) + c]
                 * cw[c * 9 + (dh + 1) * 3 + (dw + 1)];
        }
    }
    float v = acc / (1.0f + __expf(-acc));   // SiLU

    int lwh = w0 * HDIM + h0;
    size_t base = (size_t)b * KDIRS * DINNER * LTOK;
    xs[base + ((size_t)(0 * DINNER + c)) * LTOK + l]               = v;
    xs[base + ((size_t)(1 * DINNER + c)) * LTOK + lwh]             = v;
    xs[base + ((size_t)(2 * DINNER + c)) * LTOK + (LTOK - 1 - l)]  = v;
    xs[base + ((size_t)(3 * DINNER + c)) * LTOK + (LTOK - 1 - lwh)] = v;
}

// ---------------------------------------------------------------------------
// x_dbl[b][k][c][l] = sum_d x_proj_weight[k][c][d] * xs[b][k][d][l]
// M padded 38 -> 48 (rows >= 38 read as 0).  WMMA f32 16x16x4.
// ---------------------------------------------------------------------------
__global__ void gemm_xdbl(const float* __restrict__ xpw,
                          const float* __restrict__ xs,
                          float* __restrict__ xdbl) {
    int gtid = blockIdx.x * blockDim.x + threadIdx.x;
    int wave = gtid >> 5;
    int lane = gtid & 31;
    const int ntiles = LTOK / 16;            // 256
    if (wave >= BSZ * KDIRS * 3 * ntiles) return;
    int nt   = wave % ntiles;
    int rest = wave / ntiles;
    int mt   = rest % 3;
    int bk   = rest / 3;                     // b*KDIRS + k
    int k    = bk % KDIRS;

    int half = lane >> 4;
    int l16  = lane & 15;
    int m    = mt * 16 + l16;
    int col  = nt * 16 + l16;

    const float* Ablk = xpw + (size_t)k * 38 * DINNER;          // [38][192]
    const float* Bblk = xs + (size_t)bk * DINNER * LTOK;        // [192][L]
    bool mvalid = (m < 38);

    v8f c = {};
    for (int kk = 0; kk < DINNER; kk += 4) {
        int k0 = kk + 2 * half;
        v2f a, b;
        a[0] = mvalid ? Ablk[(size_t)m * DINNER + k0]     : 0.0f;
        a[1] = mvalid ? Ablk[(size_t)m * DINNER + k0 + 1] : 0.0f;
        b[0] = Bblk[(size_t)k0 * LTOK + col];
        b[1] = Bblk[(size_t)(k0 + 1) * LTOK + col];
        c = __builtin_amdgcn_wmma_f32_16x16x4_f32(false, a, false, b,
                                                  (short)0, c, false, false);
    }
    float* out = xdbl + (size_t)bk * CPAD * LTOK + (size_t)col;
#pragma unroll
    for (int j = 0; j < 8; ++j)
        out[(size_t)(mt * 16 + j + 8 * half) * LTOK] = c[j];
}

// ---------------------------------------------------------------------------
// Selective scan. Grid = B*K blocks, 192 threads; lane d owns one channel.
// Fuses dt-rank projection + bias + softplus; 16 states in registers.
// Writes y transposed: yT[b][k][l][d] (coalesced across d).
// ---------------------------------------------------------------------------
__global__ void scan_kernel(const float* __restrict__ xs,
                            const float* __restrict__ xdbl,
                            const float* __restrict__ dtw,
                            const float* __restrict__ dtb,
                            const float* __restrict__ A_logs,
                            const float* __restrict__ Ds,
                            float* __restrict__ yT) {
    int bk = blockIdx.x;
    int k  = bk % KDIRS;
    int d  = threadIdx.x;             // 0..191

    float Av[DSTATE];
#pragma unroll
    for (int n = 0; n < DSTATE; ++n)
        Av[n] = -__expf(A_logs[(size_t)(k * DINNER + d) * DSTATE + n]);
    float wr[DTRANK];
#pragma unroll
    for (int r = 0; r < DTRANK; ++r)
        wr[r] = dtw[(size_t)(k * DINNER + d) * DTRANK + r];
    float bias = dtb[k * DINNER + d];
    float Dval = Ds[k * DINNER + d];

    const float* up    = xs + ((size_t)bk * DINNER + d) * LTOK;
    const float* dbase = xdbl + (size_t)bk * CPAD * LTOK;
    float*       yp    = yT + (size_t)bk * LTOK * DINNER + d;

    float h[DSTATE];
#pragma unroll
    for (int n = 0; n < DSTATE; ++n) h[n] = 0.0f;

    for (int l = 0; l < LTOK; ++l) {
        __builtin_prefetch(up + l + 128, 0, 1);       // global_prefetch_b8
        float u = up[l];
        float draw = bias;
#pragma unroll
        for (int r = 0; r < DTRANK; ++r)
            draw = __builtin_fmaf(dbase[(size_t)r * LTOK + l], wr[r], draw);
        float delta = (draw > 20.0f) ? draw : log1pf(__expf(draw)); // softplus
        float du = delta * u;
        float y = 0.0f;
#pragma unroll
        for (int n = 0; n < DSTATE; ++n) {
            float Bt = dbase[(size_t)(DTRANK + n) * LTOK + l];
            float Ct = dbase[(size_t)(DTRANK + DSTATE + n) * LTOK + l];
            float dA = __expf(delta * Av[n]);
            h[n] = __builtin_fmaf(dA, h[n], du * Bt);
            y = __builtin_fmaf(h[n], Ct, y);
        }
        yp[(size_t)l * DINNER] = y + Dval * u;
    }
}

// ---------------------------------------------------------------------------
// Merge 4 directions + LayerNorm over 192 + SiLU gate.
// Grid = B*L blocks, 192 threads.
// ---------------------------------------------------------------------------
__global__ void combine_ln_gate(const float* __restrict__ yT,
                                const float* __restrict__ xz,
                                const float* __restrict__ ng,
                                const float* __restrict__ nb,
                                float* __restrict__ yg) {
    int bl = blockIdx.x;
    int b  = bl / LTOK;
    int l  = bl % LTOK;
    int d  = threadIdx.x;
    int h0 = l / WDIM, w0 = l % WDIM;
    int lwh = w0 * HDIM + h0;

    size_t base = (size_t)b * KDIRS * LTOK * DINNER;
    float y = yT[base + ((size_t)(0 * LTOK) + l) * DINNER + d]
            + yT[base + ((size_t)(2 * LTOK) + (LTOK - 1 - l)) * DINNER + d]
            + yT[base + ((size_t)(1 * LTOK) + lwh) * DINNER + d]
            + yT[base + ((size_t)(3 * LTOK) + (LTOK - 1 - lwh)) * DINNER + d];

    __shared__ float s_sum, s_sq;
    if (d == 0) { s_sum = 0.0f; s_sq = 0.0f; }
    __syncthreads();
    atomicAdd(&s_sum, y);
    atomicAdd(&s_sq, y * y);
    __syncthreads();

    float mu  = s_sum * (1.0f / DINNER);
    float var = s_sq * (1.0f / DINNER) - mu * mu;
    float yn  = (y - mu) * rsqrtf(var + 1e-5f) * ng[d] + nb[d];

    float z = xz[((size_t)(b * LTOK + l)) * (2 * DINNER) + DINNER + d];
    float gate = z / (1.0f + __expf(-z));
    yg[((size_t)(b * LTOK + l)) * DINNER + d] = yn * gate;
}

// ---------------------------------------------------------------------------
extern "C" void kernel_launch(void* const* d_in, const int* in_sizes, int n_in,
                              void* d_out, int out_size, void* d_ws, size_t ws_size,
                              hipStream_t stream) {
    const float* x       = (const float*)d_in[0];
    const float* ipw     = (const float*)d_in[1];   // [384,96]
    const float* cw      = (const float*)d_in[2];   // [192,1,3,3]
    const float* cb      = (const float*)d_in[3];   // [192]
    const float* xpw     = (const float*)d_in[4];   // [4,38,192]
    const float* dtw     = (const float*)d_in[5];   // [4,192,6]
    const float* dtb     = (const float*)d_in[6];   // [4,192]
    const float* A_logs  = (const float*)d_in[7];   // [768,16]
    const float* Ds      = (const float*)d_in[8];   // [768]
    const float* ng      = (const float*)d_in[9];   // [192]
    const float* nb      = (const float*)d_in[10];  // [192]
    const float* opw     = (const float*)d_in[11];  // [96,192]
    float* out = (float*)d_out;

    float* ws   = (float*)d_ws;
    float* xz   = ws;                                      // B*L*384
    float* xs   = xz   + (size_t)BSZ * LTOK * 2 * DINNER;  // B*4*192*L
    float* xdbl = xs   + (size_t)BSZ * KDIRS * DINNER * LTOK; // B*4*48*L
    float* yT   = xdbl + (size_t)BSZ * KDIRS * CPAD * LTOK;   // B*4*L*192
    float* yg   = yT   + (size_t)BSZ * KDIRS * LTOK * DINNER; // B*L*192

    const int M = BSZ * LTOK;   // 8192

    // 1) in-proj GEMM: [8192,96] x [96,384]
    {
        int waves = (M / 16) * (2 * DINNER / 16);
        int thr = waves * 32;
        gemm_xw<<<(thr + 255) / 256, 256, 0, stream>>>(x, ipw, xz, M, 2 * DINNER, DMODEL);
    }
    // 2) depthwise conv + SiLU + 4-direction scatter
    {
        int n = BSZ * LTOK * DINNER;
        conv_silu_scatter<<<(n + 255) / 256, 256, 0, stream>>>(xz, cw, cb, xs);
    }
    // 3) x_dbl GEMM per (b,k): [48,192] x [192,4096]
    {
        int waves = BSZ * KDIRS * 3 * (LTOK / 16);
        int thr = waves * 32;
        gemm_xdbl<<<(thr + 255) / 256, 256, 0, stream>>>(xpw, xs, xdbl);
    }
    // 4) selective scan
    scan_kernel<<<BSZ * KDIRS, DINNER, 0, stream>>>(xs, xdbl, dtw, dtb, A_logs, Ds, yT);
    // 5) merge + LayerNorm + gate
    combine_ln_gate<<<BSZ * LTOK, DINNER, 0, stream>>>(yT, xz, ng, nb, yg);
    // 6) out-proj GEMM: [8192,192] x [192,96]
    {
        int waves = (M / 16) * (DMODEL / 16);
        int thr = waves * 32;
        gemm_xw<<<(thr + 255) / 256, 256, 0, stream>>>(yg, opw, out, M, DMODEL, DINNER);
    }
}